// LCP_75204877353520
// MI455X (gfx1250) — compile-verified
//
#include <hip/hip_runtime.h>
#include <math.h>

typedef __attribute__((ext_vector_type(2))) float v2f;
typedef __attribute__((ext_vector_type(8))) float v8f;

#define NPTS 16384
#define KNBR 32
#define CIN  64
#define COUT 128
#define NB   4
#define EPS_IN 1e-5f

// ---- wave32 reductions (lane = k, K = 32 exactly one wave) ----
__device__ __forceinline__ float wave_sum32(float v) {
#pragma unroll
  for (int m = 16; m >= 1; m >>= 1) v += __shfl_xor(v, m, 32);
  return v;
}
__device__ __forceinline__ float wave_max32(float v) {
#pragma unroll
  for (int m = 16; m >= 1; m >>= 1) v = fmaxf(v, __shfl_xor(v, m, 32));
  return v;
}

// packed dot: y = sum_j w[j] . x[j]  over 8 v2f pairs (16 channels)
__device__ __forceinline__ v2f pk_dot8(const v2f* __restrict__ w, const v2f* __restrict__ x, v2f acc) {
#pragma unroll
  for (int j = 0; j < 8; ++j) acc = __builtin_elementwise_fma(w[j], x[j], acc);
  return acc;
}

// d_ws float layout:
//   [0..63]    sum1   [64..127]  sumsq1
//   [128..191] scale1 [192..255] shift1
//   [256..319] sum2   [320..383] sumsq2
//   [384..447] scale2 [448..511] shift2
// index j = b*16 + c

// ---------------- Pass 1: stats of y1 = fc1 . pts ----------------
__global__ void __launch_bounds__(256) k_stats1(
    const float* __restrict__ points, const float* __restrict__ support,
    const float* __restrict__ fc1, float* __restrict__ stats) {
  const int b = blockIdx.y;
  float s[16] = {}, q[16] = {};
  const int NK = NPTS * KNBR;
  for (int idx = blockIdx.x * blockDim.x + threadIdx.x; idx < NK;
       idx += gridDim.x * blockDim.x) {
    const int n = idx >> 5, k = idx & 31;
    float p[3];
#pragma unroll
    for (int d = 0; d < 3; ++d)
      p[d] = points[((size_t)(b * 3 + d) * NPTS + n) * KNBR + k] -
             support[(size_t)(b * 3 + d) * NPTS + n];
#pragma unroll
    for (int e = 0; e < 16; ++e) {
      float y = fc1[e * 3 + 0] * p[0] + fc1[e * 3 + 1] * p[1] + fc1[e * 3 + 2] * p[2];
      s[e] += y; q[e] += y * y;
    }
  }
  __shared__ float red[8][32];
  const int lane = threadIdx.x & 31, wid = threadIdx.x >> 5;
#pragma unroll
  for (int e = 0; e < 16; ++e) { s[e] = wave_sum32(s[e]); q[e] = wave_sum32(q[e]); }
  if (lane == 0) {
#pragma unroll
    for (int e = 0; e < 16; ++e) { red[wid][e] = s[e]; red[wid][16 + e] = q[e]; }
  }
  __syncthreads();
  if (wid == 0) {
    float v = 0.f;
#pragma unroll
    for (int w = 0; w < 8; ++w) v += red[w][lane];
    atomicAdd(stats + (lane < 16 ? 0 : 64) + b * 16 + (lane & 15), v);
  }
}

// ---------------- fold stats -> scale/shift ----------------
__global__ void k_finalize(float* __restrict__ stats, const float* __restrict__ g,
                           const float* __restrict__ bb, int sum_off, int out_off) {
  const int j = threadIdx.x;  // 64 threads: j = b*16+c
  const float cnt = (float)NPTS * (float)KNBR;
  const float mean = stats[sum_off + j] / cnt;
  const float var = stats[sum_off + 64 + j] / cnt - mean * mean;
  const float istd = rsqrtf(var + EPS_IN);
  const float sc = g[j & 15] * istd;
  stats[out_off + j] = sc;
  stats[out_off + 64 + j] = bb[j & 15] - mean * sc;
}

// ---------------- Pass 2: stats of y2 = fc2 . [mat1; mp1] ----------------
__global__ void __launch_bounds__(256) k_stats2(
    const float* __restrict__ points, const float* __restrict__ support,
    const float* __restrict__ fc1, const float* __restrict__ fc2,
    const float* __restrict__ alpha_p, const float* __restrict__ beta_p,
    float* __restrict__ stats) {
  const int b = blockIdx.y;
  const int lane = threadIdx.x & 31, wid = threadIdx.x >> 5;
  const float alpha = alpha_p[0], beta = beta_p[0];
  const float* scale1 = stats + 128 + b * 16;
  const float* shift1 = stats + 192 + b * 16;
  const v2f* fc2v = (const v2f*)fc2;   // row e: fc2v[e*16 + j] (m1 part), +8 (mp1 part)
  float s[16] = {}, q[16] = {};
  const int k = lane;
#pragma unroll 1
  for (int t = 0; t < 8; ++t) {
    const int n = blockIdx.x * 64 + wid * 8 + t;
    float p[3];
#pragma unroll
    for (int d = 0; d < 3; ++d)
      p[d] = points[((size_t)(b * 3 + d) * NPTS + n) * KNBR + k] -
             support[(size_t)(b * 3 + d) * NPTS + n];
    const float dist = sqrtf(p[0] * p[0] + p[1] * p[1] + p[2] * p[2]);
    const float dwr = 1.f / (1.f + __expf(alpha * dist - beta));
    float dws = wave_sum32(dwr);
    dws = dws + (dws == 0.f ? 1.f : 0.f) + 1e-6f;
    const float dw = dwr / dws * (float)KNBR;
    float m1[16];
#pragma unroll
    for (int e = 0; e < 16; ++e) {
      float y = fc1[e * 3 + 0] * p[0] + fc1[e * 3 + 1] * p[1] + fc1[e * 3 + 2] * p[2];
      m1[e] = fmaxf(y * scale1[e] + shift1[e], 0.f);
    }
    v2f m1v[8], mp1v[8];
#pragma unroll
    for (int j = 0; j < 8; ++j) {
      m1v[j].x = m1[2 * j];     m1v[j].y = m1[2 * j + 1];
      mp1v[j].x = wave_max32(m1[2 * j] * dw);
      mp1v[j].y = wave_max32(m1[2 * j + 1] * dw);
    }
#pragma unroll
    for (int e = 0; e < 16; ++e) {
      v2f acc = {0.f, 0.f};
      acc = pk_dot8(fc2v + e * 16, m1v, acc);
      acc = pk_dot8(fc2v + e * 16 + 8, mp1v, acc);
      const float y = acc.x + acc.y;
      s[e] += y; q[e] += y * y;
    }
  }
  __shared__ float red[8][32];
#pragma unroll
  for (int e = 0; e < 16; ++e) { s[e] = wave_sum32(s[e]); q[e] = wave_sum32(q[e]); }
  if (lane == 0) {
#pragma unroll
    for (int e = 0; e < 16; ++e) { red[wid][e] = s[e]; red[wid][16 + e] = q[e]; }
  }
  __syncthreads();
  if (wid == 0) {
    float v = 0.f;
#pragma unroll
    for (int w = 0; w < 8; ++w) v += red[w][lane];
    atomicAdd(stats + (lane < 16 ? 256 : 320) + b * 16 + (lane & 15), v);
  }
}

// ---------------- Main fused kernel ----------------
// block = 256 (8 waves); one workgroup handles (b, 16 n's).
// LDS: mat3LDS[16n][32k][16e] (8192 f) + featLDS[1024][17 pad] (17408 f)
__global__ void __launch_bounds__(256) k_main(
    const float* __restrict__ input, const float* __restrict__ points,
    const float* __restrict__ support, const float* __restrict__ fc1,
    const float* __restrict__ fc2, const float* __restrict__ fc3,
    const float* __restrict__ alpha_p, const float* __restrict__ beta_p,
    const float* __restrict__ stats, const float* __restrict__ cvw,
    const float* __restrict__ cvb, float* __restrict__ out) {
  extern __shared__ float lds[];
  float* mat3LDS = lds;                 // [n_loc][k][e]
  float* featLDS = lds + 16 * 32 * 16;  // [kc=0..1023][n_loc], stride 17

  const int b = blockIdx.y;
  const int n0 = blockIdx.x * 16;
  const int lane = threadIdx.x & 31;
  const int wid = threadIdx.x >> 5;
  const float alpha = alpha_p[0], beta = beta_p[0];
  const float* scale1 = stats + 128 + b * 16;
  const float* shift1 = stats + 192 + b * 16;
  const float* scale2 = stats + 384 + b * 16;
  const float* shift2 = stats + 448 + b * 16;
  const v2f* fc2v = (const v2f*)fc2;
  const v2f* fc3v = (const v2f*)fc3;

  // ---- Phase 1: per-point MLP -> mat3 in LDS (one wave per n, lane = k) ----
#pragma unroll 1
  for (int t = 0; t < 2; ++t) {
    const int n_loc = wid * 2 + t;
    const int n = n0 + n_loc;
    const int k = lane;
    float p[3];
#pragma unroll
    for (int d = 0; d < 3; ++d)
      p[d] = points[((size_t)(b * 3 + d) * NPTS + n) * KNBR + k] -
             support[(size_t)(b * 3 + d) * NPTS + n];
    const float dist = sqrtf(p[0] * p[0] + p[1] * p[1] + p[2] * p[2]);
    const float dwr = 1.f / (1.f + __expf(alpha * dist - beta));
    float dws = wave_sum32(dwr);
    dws = dws + (dws == 0.f ? 1.f : 0.f) + 1e-6f;
    const float dw = dwr / dws * (float)KNBR;

    float m1[16], m2[16];
#pragma unroll
    for (int e = 0; e < 16; ++e) {
      float y = fc1[e * 3 + 0] * p[0] + fc1[e * 3 + 1] * p[1] + fc1[e * 3 + 2] * p[2];
      m1[e] = fmaxf(y * scale1[e] + shift1[e], 0.f);
    }
    v2f m1v[8], mp1v[8];
#pragma unroll
    for (int j = 0; j < 8; ++j) {
      m1v[j].x = m1[2 * j];     m1v[j].y = m1[2 * j + 1];
      mp1v[j].x = wave_max32(m1[2 * j] * dw);
      mp1v[j].y = wave_max32(m1[2 * j + 1] * dw);
    }
#pragma unroll
    for (int e = 0; e < 16; ++e) {
      v2f acc = {0.f, 0.f};
      acc = pk_dot8(fc2v + e * 16, m1v, acc);
      acc = pk_dot8(fc2v + e * 16 + 8, mp1v, acc);
      const float y = acc.x + acc.y;
      m2[e] = fmaxf(y * scale2[e] + shift2[e], 0.f);
    }
    v2f m2v[8], mp2v[8];
#pragma unroll
    for (int j = 0; j < 8; ++j) {
      m2v[j].x = m2[2 * j];     m2v[j].y = m2[2 * j + 1];
      mp2v[j].x = wave_max32(m2[2 * j] * dw);
      mp2v[j].y = wave_max32(m2[2 * j + 1] * dw);
    }
    float* dst = mat3LDS + ((n_loc * KNBR + k) * 16);
#pragma unroll
    for (int e = 0; e < 16; ++e) {
      v2f acc = {0.f, 0.f};
      acc = pk_dot8(fc3v + e * 16, m2v, acc);
      acc = pk_dot8(fc3v + e * 16 + 8, mp2v, acc);
      dst[e] = fmaxf(acc.x + acc.y, 0.f) * dw;   // mat3[e] for this (n,k)
    }
  }
  __syncthreads();

  // ---- Phase 2 (WMMA): feat[i,e] = sum_k input[i,k] * mat3[e,k] per n ----
  const int hi = lane >> 4;       // half-wave select
  const int row = lane & 15;      // A row / B-C column
  const size_t istride = (size_t)NPTS * KNBR;
#pragma unroll 1
  for (int jj = 0; jj < 8; ++jj) {
    const int job = wid * 8 + jj;        // 64 jobs = 16 n * 4 i-tiles
    const int n_loc = job >> 2;
    const int it = job & 3;
    const int n = n0 + n_loc;
    const float* A0 = input + (size_t)b * CIN * istride + (size_t)n * KNBR +
                      (size_t)(it * 16 + row) * istride + 2 * hi;
    v8f acc = {};
#pragma unroll
    for (int s = 0; s < 8; ++s) {        // K = 32 in steps of 4
      v2f af = *(const v2f*)(A0 + s * 4);                       // A[row, k0|k0+2 .. +1]
      const float* bp = mat3LDS + (n_loc * 32 + s * 4 + 2 * hi) * 16 + row;
      v2f bf; bf.x = bp[0]; bf.y = bp[16];                      // B[k, e=row]
      acc = __builtin_amdgcn_wmma_f32_16x16x4_f32(
          false, af, false, bf, (short)0, acc, false, false);
    }
#pragma unroll
    for (int v = 0; v < 8; ++v) {
      const int kc = (it * 16 + v + 8 * hi) * 16 + row;  // kc = i*16 + e
      featLDS[kc * 17 + n_loc] = acc[v];
    }
  }
  __syncthreads();

  // ---- Phase 3 (WMMA): out[o, n] = cv_w[o, 0..1023] . feat[0..1023, n] ----
  {
    const float* Wr = cvw + (size_t)(wid * 16 + row) * 1024 + 2 * hi;
    v8f acc = {};
#pragma unroll 4
    for (int s = 0; s < 256; ++s) {      // K = 1024 in steps of 4
      v2f af = *(const v2f*)(Wr + s * 4);
      const float* bp = featLDS + (s * 4 + 2 * hi) * 17 + row;  // col n = row
      v2f bf; bf.x = bp[0]; bf.y = bp[17];
      acc = __builtin_amdgcn_wmma_f32_16x16x4_f32(
          false, af, false, bf, (short)0, acc, false, false);
    }
    const int ncol = n0 + row;
#pragma unroll
    for (int v = 0; v < 8; ++v) {
      const int o = wid * 16 + v + 8 * hi;
      out[((size_t)b * COUT + o) * NPTS + ncol] = acc[v] + cvb[o];
    }
  }
}

// ---------------- copy support_points into output tail ----------------
__global__ void k_copy(const float* __restrict__ src, float* __restrict__ dst, int n) {
  const int i = blockIdx.x * blockDim.x + threadIdx.x;
  if (i < n) dst[i] = src[i];
}

extern "C" void kernel_launch(void* const* d_in, const int* in_sizes, int n_in,
                              void* d_out, int out_size, void* d_ws, size_t ws_size,
                              hipStream_t stream) {
  const float* input   = (const float*)d_in[0];
  const float* points  = (const float*)d_in[1];
  const float* support = (const float*)d_in[2];
  const float* fc1     = (const float*)d_in[3];
  const float* fc2     = (const float*)d_in[4];
  const float* fc3     = (const float*)d_in[5];
  const float* bn1g    = (const float*)d_in[6];
  const float* bn1b    = (const float*)d_in[7];
  const float* bn2g    = (const float*)d_in[8];
  const float* bn2b    = (const float*)d_in[9];
  const float* alpha   = (const float*)d_in[10];
  const float* beta    = (const float*)d_in[11];
  const float* cvw     = (const float*)d_in[12];
  const float* cvb     = (const float*)d_in[13];
  float* out = (float*)d_out;
  float* stats = (float*)d_ws;

  hipMemsetAsync(stats, 0, 512 * sizeof(float), stream);
  k_stats1<<<dim3(256, NB), 256, 0, stream>>>(points, support, fc1, stats);
  k_finalize<<<1, 64, 0, stream>>>(stats, bn1g, bn1b, 0, 128);
  k_stats2<<<dim3(256, NB), 256, 0, stream>>>(points, support, fc1, fc2, alpha, beta, stats);
  k_finalize<<<1, 64, 0, stream>>>(stats, bn2g, bn2b, 256, 384);

  const size_t ldsBytes = (size_t)(16 * 32 * 16 + 1024 * 17) * sizeof(float);
  k_main<<<dim3(NPTS / 16, NB), 256, ldsBytes, stream>>>(
      input, points, support, fc1, fc2, fc3, alpha, beta, stats, cvw, cvb, out);

  k_copy<<<(NB * 3 * NPTS + 255) / 256, 256, 0, stream>>>(
      support, out + (size_t)NB * COUT * NPTS, NB * 3 * NPTS);
}